// Sinkhorn_47029891891605
// MI455X (gfx1250) — compile-verified
//
#include <hip/hip_runtime.h>

// ---------------------------------------------------------------------------
// Sinkhorn (log-domain, potential-vector formulation) for MI455X / gfx1250.
//   y0 = x / TAU;  f[i] = LSE_j(y0 - g[j]);  g[j] = LSE_i(y0 - f[i]);  x5 iters
//   out = exp(y0 - f - g) + EPS
// Bandwidth-bound: 12 passes over 256MB ~= 3.1 GB -> ~130us at 23.3 TB/s.
// Row pass staged through LDS by the Tensor Data Mover (double buffered).
// No matmul exists in this op => no WMMA; the gfx1250 win is TDM + L2 reuse.
// ---------------------------------------------------------------------------

#define INV_TAU   100.0f
#define EPS_ADD   1e-6f
#define NEG_BIG   (-3.402823466e38f)

typedef unsigned int u32;
typedef unsigned long long u64;
typedef __attribute__((ext_vector_type(4))) u32   v4u;
typedef __attribute__((ext_vector_type(8))) int   v8i;
typedef __attribute__((ext_vector_type(4))) int   v4i;
typedef __attribute__((ext_vector_type(4))) float f32x4;

// --- Tensor Data Mover: linear 4KB row copy, global -> LDS ------------------
// D# built per CDNA5 ISA ch.8: data_size=3 (8B units), 1D tile of 512 units.
// This toolchain exposes the 6-arg builtin:
//   (uint32x4 g0, int32x8 g1, int32x4, int32x4, int32x8, i32 cpol)
__device__ __forceinline__ void tdm_load_row(const void* gptr, u32 lds_byte_off) {
  u64 ga = (u64)gptr;
  v4u g0;
  g0[0] = 1u;                                   // count=1, user mode, no gather
  g0[1] = lds_byte_off;                         // lds_addr (bytes)
  g0[2] = (u32)(ga & 0xFFFFFFFFu);              // global_addr[31:0]
  g0[3] = (u32)((ga >> 32) & 0x01FFFFFFu)       // global_addr[56:32]
        | 0x80000000u;                          // type=2 ("image")
  v8i g1;
  g1[0] = 0x00030000;                           // wg_mask=0, data_size=3 (8B)
  g1[1] = (int)(512u << 16);                    // tensor_dim0 = 512 (lo16)
  g1[2] = (int)(1u << 16);                      // tensor_dim0 hi=0, tensor_dim1=1
  g1[3] = (int)(512u << 16);                    // tensor_dim1 hi=0, tile_dim0=512
  g1[4] = 1;                                    // tile_dim1=1, tile_dim2=0
  g1[5] = 512;                                  // tensor_dim0_stride lo32
  g1[6] = (int)(512u << 16);                    // dim0_stride hi=0, dim1_stride lo16
  g1[7] = 0;                                    // dim1_stride hi
  v4i z4 = {0, 0, 0, 0};
  v8i z8 = {0, 0, 0, 0, 0, 0, 0, 0};
  __builtin_amdgcn_tensor_load_to_lds(g0, g1, z4, z4, z8, 0);
}

// --- Row pass: f[b,i] = LSE_j( x[b,i,j]*INV_TAU - g[b,j] ) ------------------
// block = 128 thr (4 waves); each wave: 8 rows, TDM double-buffered via LDS.
__global__ __launch_bounds__(128) void rowpass(const float* __restrict__ x,
                                               const float* __restrict__ g,
                                               float* __restrict__ f) {
  __shared__ float sh_g[1024];
  __shared__ float sh_buf[4][2][1024];          // 4 waves x 2 x 4KB
  const int b    = blockIdx.y;
  const int row0 = blockIdx.x * 32;
  const int tid  = threadIdx.x;
  const int wave = tid >> 5;
  const int lane = tid & 31;

  {                                             // stage g: 2x float4 per thread
    const f32x4* gsrc = (const f32x4*)(g + ((size_t)b << 10));
    f32x4* gdst = (f32x4*)sh_g;
    gdst[tid]       = gsrc[tid];
    gdst[tid + 128] = gsrc[tid + 128];
  }
  __syncthreads();

  const f32x4* gv4 = (const f32x4*)sh_g;
  const float* xw  = x + (((size_t)b << 20) + ((size_t)(row0 + wave * 8) << 10));
  const u32 lds0 = (u32)(size_t)&sh_buf[wave][0][0];
  const u32 lds1 = (u32)(size_t)&sh_buf[wave][1][0];

  tdm_load_row(xw, lds0);                       // prime buffer 0
  for (int r = 0; r < 8; ++r) {
    if (r < 7) {
      tdm_load_row(xw + ((size_t)(r + 1) << 10), (r & 1) ? lds0 : lds1);
      __builtin_amdgcn_s_wait_tensorcnt(1);     // oldest (current row) done
    } else {
      __builtin_amdgcn_s_wait_tensorcnt(0);
    }
    asm volatile("" ::: "memory");

    const f32x4* bv = (const f32x4*)&sh_buf[wave][r & 1][0];
    float v[32];
    float m = NEG_BIG;
#pragma unroll
    for (int q = 0; q < 8; ++q) {               // lane-consecutive 16B: no conflicts
      f32x4 xv = bv[q * 32 + lane];
      f32x4 gg = gv4[q * 32 + lane];
      v[4 * q + 0] = xv.x * INV_TAU - gg.x;
      v[4 * q + 1] = xv.y * INV_TAU - gg.y;
      v[4 * q + 2] = xv.z * INV_TAU - gg.z;
      v[4 * q + 3] = xv.w * INV_TAU - gg.w;
      m = fmaxf(m, fmaxf(fmaxf(v[4 * q], v[4 * q + 1]),
                         fmaxf(v[4 * q + 2], v[4 * q + 3])));
    }
#pragma unroll
    for (int off = 16; off; off >>= 1) m = fmaxf(m, __shfl_xor(m, off, 32));
    float s = 0.0f;
#pragma unroll
    for (int k = 0; k < 32; ++k) s += __expf(v[k] - m);
#pragma unroll
    for (int off = 16; off; off >>= 1) s += __shfl_xor(s, off, 32);
    if (lane == 0) f[(b << 10) + row0 + wave * 8 + r] = m + __logf(s);
  }
}

// --- Col pass (partials): one column/thread, 128-row chunk, online LSE ------
__global__ __launch_bounds__(256) void colpart(const float* __restrict__ x,
                                               const float* __restrict__ f,
                                               float* __restrict__ mp,
                                               float* __restrict__ sp) {
  __shared__ float shf[128];
  const int j  = blockIdx.x * 256 + threadIdx.x;
  const int rc = blockIdx.y;                    // row chunk 0..7
  const int b  = blockIdx.z;
  if (threadIdx.x < 32) {                       // stage f chunk: one wave, float4
    const f32x4* fsrc = (const f32x4*)(f + ((size_t)(b << 10) + rc * 128));
    ((f32x4*)shf)[threadIdx.x] = fsrc[threadIdx.x];
  }
  __syncthreads();

  const float* p = x + (((size_t)b << 20) + ((size_t)(rc * 128) << 10) + j);
  float m = NEG_BIG, s = 0.0f;
#pragma unroll 8
  for (int r = 0; r < 128; ++r) {               // coalesced 1KB/row/block
    float v  = p[(size_t)r << 10] * INV_TAU - shf[r];
    float nm = fmaxf(m, v);
    s = s * __expf(m - nm) + __expf(v - nm);    // branch-free online LSE
    m = nm;
  }
  const int o = (((b << 3) + rc) << 10) + j;
  mp[o] = m;
  sp[o] = s;
}

// --- Merge 8 chunk partials -> g --------------------------------------------
__global__ __launch_bounds__(1024) void colmerge(const float* __restrict__ mp,
                                                 const float* __restrict__ sp,
                                                 float* __restrict__ g) {
  const int b = blockIdx.x;
  const int j = threadIdx.x;
  float m = NEG_BIG, s = 0.0f;
#pragma unroll
  for (int c = 0; c < 8; ++c) {
    const int o = (((b << 3) + c) << 10) + j;
    float mm = mp[o], ss = sp[o];
    float nm = fmaxf(m, mm);
    s = s * __expf(m - nm) + ss * __expf(mm - nm);
    m = nm;
  }
  g[(b << 10) + j] = m + __logf(s);
}

// --- Final: out = exp(y0 - f - g) + EPS. NT load (last use of x) + NT store -
__global__ __launch_bounds__(256) void finalpass(const float* __restrict__ x,
                                                 const float* __restrict__ f,
                                                 const float* __restrict__ g,
                                                 float* __restrict__ out) {
  const int b = blockIdx.y;
  const int i = blockIdx.x;
  const int c = threadIdx.x;                    // float4 index 0..255
  const size_t base = ((size_t)b << 20) + ((size_t)i << 10);
  const float fi = f[(b << 10) + i];            // uniform -> scalar load
  const f32x4 gv = ((const f32x4*)(g + ((size_t)b << 10)))[c];
  const f32x4 xv = __builtin_nontemporal_load((const f32x4*)(x + base) + c);
  f32x4 o;
  o.x = __expf(xv.x * INV_TAU - fi - gv.x) + EPS_ADD;
  o.y = __expf(xv.y * INV_TAU - fi - gv.y) + EPS_ADD;
  o.z = __expf(xv.z * INV_TAU - fi - gv.z) + EPS_ADD;
  o.w = __expf(xv.w * INV_TAU - fi - gv.w) + EPS_ADD;
  __builtin_nontemporal_store(o, (f32x4*)(out + base) + c);
}

// ---------------------------------------------------------------------------
extern "C" void kernel_launch(void* const* d_in, const int* in_sizes, int n_in,
                              void* d_out, int out_size, void* d_ws, size_t ws_size,
                              hipStream_t stream) {
  const float* x = (const float*)d_in[0];
  float* out = (float*)d_out;

  float* f  = (float*)d_ws;                     // 64*1024
  float* g  = f + 65536;                        // 64*1024
  float* mp = g + 65536;                        // 64*8*1024
  float* sp = mp + 524288;                      // 64*8*1024   (total ~4.5MB)

  (void)hipMemsetAsync(g, 0, 65536 * sizeof(float), stream);  // g0 = 0

  for (int it = 0; it < 5; ++it) {
    rowpass<<<dim3(32, 64), 128, 0, stream>>>(x, g, f);
    colpart<<<dim3(4, 8, 64), 256, 0, stream>>>(x, f, mp, sp);
    colmerge<<<dim3(64), 1024, 0, stream>>>(mp, sp, g);
  }
  finalpass<<<dim3(1024, 64), 256, 0, stream>>>(x, f, g, out);
}